// ConcatMLPAggregator_10230612099227
// MI455X (gfx1250) — compile-verified
//
#include <hip/hip_runtime.h>
#include <math.h>

typedef __attribute__((ext_vector_type(2))) float v2f;
typedef __attribute__((ext_vector_type(8))) float v8f;

#define L_SEQ   64     // sequence length
#define KSET    8      // MAX_SET_SIZE
#define DV      128    // d_v
#define HID     128    // hidden
#define M_TILE  16     // chains per block (WMMA M)
#define HSTR    132    // LDS stride for h tile (avoid bank conflicts)

#define W1_PAIRS 65536   // 256 k4-steps * 2 halves * 128 cols
#define W2_PAIRS 8192    // 32 k4-steps * 2 halves * 128 cols
#define ZP_FLOATS 128

// Repack W1/W2 into WMMA B-fragment order and write the zero page.
// Bpack[((k4*2+hi)*128+col)*2 + {0,1}] = W[(4*k4+2*hi+{0,1})*128 + col]
__global__ __launch_bounds__(256) void prep_pack_kernel(
    const float* __restrict__ W1, const float* __restrict__ W2,
    float* __restrict__ bpack1, float* __restrict__ bpack2,
    float* __restrict__ zeropage)
{
    const int p = blockIdx.x * blockDim.x + threadIdx.x;
    if (p < W1_PAIRS) {
        const int k4 = p >> 8, hi = (p >> 7) & 1, col = p & 127;
        const int kk = 4 * k4 + 2 * hi;
        bpack1[2 * p]     = W1[(size_t)kk * HID + col];
        bpack1[2 * p + 1] = W1[(size_t)(kk + 1) * HID + col];
    } else if (p < W1_PAIRS + W2_PAIRS) {
        const int q = p - W1_PAIRS;
        const int k4 = q >> 8, hi = (q >> 7) & 1, col = q & 127;
        const int kk = 4 * k4 + 2 * hi;
        bpack2[2 * q]     = W2[(size_t)kk * DV + col];
        bpack2[2 * q + 1] = W2[(size_t)(kk + 1) * DV + col];
    } else if (p < W1_PAIRS + W2_PAIRS + ZP_FLOATS) {
        zeropage[p - W1_PAIRS - W2_PAIRS] = 0.0f;
    }
}

template <bool PACKED>
__global__ __launch_bounds__(256) void concat_mlp_aggregator_kernel(
    const float* __restrict__ v,            // (8192, 64, 128)
    const int* __restrict__ batch_idx,      // (n_chains)
    const unsigned char* __restrict__ mask, // (n_chains, 64) bool bytes
    const int* __restrict__ count,          // (n_chains)
    const float* __restrict__ W1,           // (1025, 128) row-major
    const float* __restrict__ b1,           // (128)
    const float* __restrict__ W2,           // (128, 128) row-major
    const float* __restrict__ b2,           // (128)
    const float* __restrict__ bpack1,       // packed W1 fragments (or null)
    const float* __restrict__ bpack2,       // packed W2 fragments (or null)
    long long zp_off,                       // (zeropage - v) in elements
    float* __restrict__ out)                // (n_chains, 128)
{
    __shared__ long long    s_aoff[M_TILE][KSET];   // gather offsets vs v (packed)
    __shared__ unsigned int s_srcoff[M_TILE][KSET]; // element offsets (fallback)
    __shared__ float        s_scale[M_TILE][KSET];  // 0/1 slot scale (fallback)
    __shared__ float        s_lc[M_TILE];           // log1p(count)
    __shared__ float        s_h[M_TILE * HSTR];     // gelu hidden tile

    const int tid    = threadIdx.x;
    const int chain0 = blockIdx.x * M_TILE;

    // ---------------- Phase 0: decode masks, build gather table ----------------
    if (tid < M_TILE) {
        const int chain = chain0 + tid;
        const unsigned char* mrow = mask + (size_t)chain * L_SEQ;
        unsigned long long bits = 0ull;
        #pragma unroll
        for (int c8 = 0; c8 < 8; ++c8) {
            unsigned long long chunk =
                *reinterpret_cast<const unsigned long long*>(mrow + c8 * 8);
            #pragma unroll
            for (int b = 0; b < 8; ++b)
                bits |= ((chunk >> (8 * b)) & 1ull) << (c8 * 8 + b);
        }
        const unsigned int vrow = (unsigned int)batch_idx[chain];
        #pragma unroll
        for (int s = 0; s < KSET; ++s) {
            unsigned int pos = 0u;
            bool picked = false;
            if (bits) {                       // first-8 set bits, ascending
                pos = (unsigned int)__builtin_ctzll(bits);
                bits &= bits - 1ull;
                picked = true;
            }
            if (PACKED) {
                s_aoff[tid][s] = picked
                    ? (long long)((unsigned long long)vrow * (L_SEQ * DV)
                                  + (unsigned long long)pos * DV)
                    : zp_off;                 // zero page lives in d_ws
            } else {
                s_srcoff[tid][s] = vrow * (unsigned int)(L_SEQ * DV) + pos * (unsigned int)DV;
                s_scale[tid][s]  = picked ? 1.0f : 0.0f;
            }
        }
        s_lc[tid] = log1pf((float)count[chain]);
    }
    __syncthreads();

    const int wave = tid >> 5;
    const int lane = tid & 31;
    const int row  = lane & 15;        // WMMA A row / M within half
    const int hi   = lane >> 4;        // lane-half selector
    const int col  = wave * 16 + row;  // this lane's output column (N)
    const int koff = 2 * hi;           // K offset of lane-half within 4-wide step

    // ---------------- Phase 1: GEMM1  x(16x1024) @ W1(1024x128) ---------------
    v8f acc = {0.f, 0.f, 0.f, 0.f, 0.f, 0.f, 0.f, 0.f};

    if (PACKED) {
        const float* __restrict__ bp = bpack1 + (size_t)(hi * 128 + col) * 2;
        for (int slot = 0; slot < KSET; ++slot) {
            // base(v) + 64-bit offset keeps these as GLOBAL loads (not FLAT)
            const float* __restrict__ ap = v + s_aoff[row][slot] + koff;
            const float* __restrict__ wp = bp + (size_t)(slot * 32) * 512;
            #pragma unroll 8
            for (int t = 0; t < 32; ++t) {
                v2f a = *reinterpret_cast<const v2f*>(ap + t * 4);
                v2f b = *reinterpret_cast<const v2f*>(wp + (size_t)t * 512);
                acc = __builtin_amdgcn_wmma_f32_16x16x4_f32(
                    false, a, false, b, (short)0, acc, false, false);
            }
        }
    } else {
        for (int slot = 0; slot < KSET; ++slot) {
            const float sc = s_scale[row][slot];
            const float* __restrict__ vptr = v + s_srcoff[row][slot] + koff;
            const float* __restrict__ wptr = W1 + (size_t)(slot * DV + koff) * HID + col;
            #pragma unroll 8
            for (int t = 0; t < 32; ++t) {
                v2f a = *reinterpret_cast<const v2f*>(vptr + t * 4);
                a.x *= sc; a.y *= sc;
                v2f b;
                b.x = wptr[(size_t)(t * 4) * HID];
                b.y = wptr[(size_t)(t * 4 + 1) * HID];
                acc = __builtin_amdgcn_wmma_f32_16x16x4_f32(
                    false, a, false, b, (short)0, acc, false, false);
            }
        }
    }

    // epilogue: k=1024 (log1p(count) column) + bias, exact gelu, stage h in LDS
    {
        const float w1last = W1[(size_t)(KSET * DV) * HID + col]; // row 1024
        const float bb1    = b1[col];
        #pragma unroll
        for (int r = 0; r < 8; ++r) {
            const int m = r + 8 * hi;               // C/D layout: M = r + 8*half
            float x = acc[r] + s_lc[m] * w1last + bb1;
            float g = 0.5f * x * (1.0f + erff(x * 0.70710678118654752f));
            s_h[m * HSTR + col] = g;
        }
    }
    __syncthreads();

    // ---------------- Phase 2: GEMM2  h(16x128) @ W2(128x128) -----------------
    v8f acc2 = {0.f, 0.f, 0.f, 0.f, 0.f, 0.f, 0.f, 0.f};
    {
        const float* __restrict__ hrow = &s_h[row * HSTR + koff];
        if (PACKED) {
            const float* __restrict__ bp2 = bpack2 + (size_t)(hi * 128 + col) * 2;
            #pragma unroll 8
            for (int t = 0; t < 32; ++t) {
                v2f a = *reinterpret_cast<const v2f*>(hrow + t * 4);   // ds_load_b64
                v2f b = *reinterpret_cast<const v2f*>(bp2 + (size_t)t * 512);
                acc2 = __builtin_amdgcn_wmma_f32_16x16x4_f32(
                    false, a, false, b, (short)0, acc2, false, false);
            }
        } else {
            const float* __restrict__ wptr = W2 + (size_t)koff * DV + col;
            #pragma unroll 8
            for (int t = 0; t < 32; ++t) {
                v2f a = *reinterpret_cast<const v2f*>(hrow + t * 4);
                v2f b;
                b.x = wptr[(size_t)(t * 4) * DV];
                b.y = wptr[(size_t)(t * 4 + 1) * DV];
                acc2 = __builtin_amdgcn_wmma_f32_16x16x4_f32(
                    false, a, false, b, (short)0, acc2, false, false);
            }
        }
    }

    const float bb2 = b2[col];
    #pragma unroll
    for (int r = 0; r < 8; ++r) {
        const int m = r + 8 * hi;
        out[(size_t)(chain0 + m) * DV + col] = acc2[r] + bb2;
    }
}

extern "C" void kernel_launch(void* const* d_in, const int* in_sizes, int n_in,
                              void* d_out, int out_size, void* d_ws, size_t ws_size,
                              hipStream_t stream) {
    const float* v            = (const float*)d_in[0];
    const int* batch_idx      = (const int*)d_in[1];
    const unsigned char* mask = (const unsigned char*)d_in[2];
    const int* count          = (const int*)d_in[3];
    const float* W1           = (const float*)d_in[4];
    const float* b1           = (const float*)d_in[5];
    const float* W2           = (const float*)d_in[6];
    const float* b2           = (const float*)d_in[7];
    float* out                = (float*)d_out;

    const int n_chains = in_sizes[3];           // count: one per chain
    dim3 grid(n_chains / M_TILE), block(256);

    const size_t need = (size_t)(W1_PAIRS * 2 + W2_PAIRS * 2 + ZP_FLOATS) * sizeof(float);
    if (ws_size >= need) {
        float* bpack1   = (float*)d_ws;
        float* bpack2   = bpack1 + (size_t)W1_PAIRS * 2;
        float* zeropage = bpack2 + (size_t)W2_PAIRS * 2;
        const long long zp_off = (long long)(zeropage - v);  // element offset vs v
        const int prep_threads = W1_PAIRS + W2_PAIRS + ZP_FLOATS;
        hipLaunchKernelGGL(prep_pack_kernel, dim3((prep_threads + 255) / 256), dim3(256),
                           0, stream, W1, W2, bpack1, bpack2, zeropage);
        hipLaunchKernelGGL((concat_mlp_aggregator_kernel<true>), grid, block, 0, stream,
                           v, batch_idx, mask, count, W1, b1, W2, b2,
                           bpack1, bpack2, zp_off, out);
    } else {
        hipLaunchKernelGGL((concat_mlp_aggregator_kernel<false>), grid, block, 0, stream,
                           v, batch_idx, mask, count, W1, b1, W2, b2,
                           (const float*)nullptr, (const float*)nullptr,
                           (long long)0, out);
    }
    (void)n_in; (void)out_size;
}